// convModel_82016695484587
// MI455X (gfx1250) — compile-verified
//
#include <hip/hip_runtime.h>

#define B_ROWS 8192
#define NZ     2500
#define NREFL  2499
#define NT_NEW 2000
#define NT_W   101
#define DZf    2.5f
#define DTf    0.001f

#define ROW_TILE 16
#define PAD_L    128           // zero pad covering j in [-128,-1]
#define RSTRIDE  2132          // 2128 needed + pad; 2132 % 64 == 20 -> no LDS bank conflicts over 16 rows
#define WEXT_OFF 16            // wext[k] for k in [-16,127] stored at wl[16+k]
#define REFL_OFF 160           // float offset of refl image inside dynamic LDS
#define SMEM_FLOATS (REFL_OFF + ROW_TILE * RSTRIDE)

typedef float v2f __attribute__((ext_vector_type(2)));
typedef float v8f __attribute__((ext_vector_type(8)));

__global__ __launch_bounds__(256)
void convmodel_wmma_kernel(const float* __restrict__ vmodel,
                           const float* __restrict__ wavelet,
                           float* __restrict__ out) {
  extern __shared__ float smem[];
  float* wl   = smem;                 // extended wavelet, zero outside [0,100]
  float* refl = smem + REFL_OFF;      // 16 x RSTRIDE sparse reflectivity image

  const int tid  = threadIdx.x;
  const int lane = tid & 31;
  const int wave = tid >> 5;          // 0..7
  const int row0 = blockIdx.x * ROW_TILE;

  // ---- zero LDS (refl image + wavelet pad) ----
  for (int i = tid; i < SMEM_FLOATS; i += 256) smem[i] = 0.0f;
  __syncthreads();

  if (tid < NT_W) wl[WEXT_OFF + tid] = wavelet[tid];

  // ---- Phase A: scan + scatter. 2 rows per wave, 16-lane segments ----
  const int seg  = lane & 15;
  const int half = lane >> 4;
  const int mrow = 2 * wave + half;   // local row 0..15
  const float* vrow = vmodel + (size_t)(row0 + mrow) * NZ;

  float carry = 0.0f;
  for (int base = 0; base < NREFL; base += 16) {
    const int i = base + seg;
    const bool valid = (i < NREFL);
    const float v0 = valid ? vrow[i]     : 1.0f;
    const float v1 = valid ? vrow[i + 1] : 1.0f;
    float x = valid ? (DZf / v0) : 0.0f;          // dt contribution
    // 16-lane inclusive scan
    #pragma unroll
    for (int d = 1; d < 16; d <<= 1) {
      float y = __shfl_up(x, d, 16);
      if (seg >= d) x += y;
    }
    const float tI = carry + x;
    carry += __shfl(x, 15, 16);                   // segment total
    const int idx  = __float2int_rn(tI / DTf);    // matches jnp.round(time/DT)
    const int idxn = __shfl_down(idx, 1, 16);     // next lane's idx (clamped at seg end)
    const bool last = (seg == 15) || (idxn != idx) || (i == NREFL - 1);
    if (valid && last && idx >= 0 && idx < NT_NEW)
      refl[mrow * RSTRIDE + PAD_L + idx] = (v1 - v0) / (v1 + v0);
    if (__all(carry > 1.9996f)) break;            // uniform per wave: all future idx >= 2000
  }
  __syncthreads();

  // ---- Phase B: FIR as GEMM via V_WMMA_F32_16X16X4_F32 ----
  const int n  = lane & 15;           // N (time within tile) / M for A loads
  const int kh = lane >> 4;           // which K-pair this half-wave holds

  // Toeplitz wavelet fragments: B[k,n] = wext[112 - 4c + n - k], identical for every tile
  v2f bfrag[29];
  #pragma unroll
  for (int c = 3; c < 32; ++c) {
    const int a0 = 112 - 4 * c + n - 2 * kh;
    bfrag[c - 3].x = wl[WEXT_OFF + a0];       // k = 2*kh
    bfrag[c - 3].y = wl[WEXT_OFF + a0 - 1];   // k = 2*kh + 1
  }

  const int m = lane & 15;            // A-matrix row = local batch row
  for (int tile = wave; tile < NT_NEW / 16; tile += 8) {
    const int t0 = tile * 16;
    v8f acc = {0.f, 0.f, 0.f, 0.f, 0.f, 0.f, 0.f, 0.f};
    // LDS column for chunk c: 128 + (t0 - 112 + 4c) + 2*kh  (8-byte aligned)
    const float* arow = refl + m * RSTRIDE + 16 + t0 + 2 * kh;
    #pragma unroll
    for (int c = 3; c < 32; ++c) {
      v2f a = *(const v2f*)(arow + 4 * c);
      acc = __builtin_amdgcn_wmma_f32_16x16x4_f32(
          /*neg_a=*/false, a, /*neg_b=*/false, bfrag[c - 3],
          /*c_mod=*/(short)0, acc, /*reuse_a=*/false, /*reuse_b=*/false);
    }
    // D layout: VGPR g -> M = g + 8*kh, N = lane&15
    float* orow = out + (size_t)(row0 + 8 * kh) * NT_NEW + t0 + n;
    #pragma unroll
    for (int g = 0; g < 8; ++g)
      orow[(size_t)g * NT_NEW] = acc[g];
  }
}

extern "C" void kernel_launch(void* const* d_in, const int* in_sizes, int n_in,
                              void* d_out, int out_size, void* d_ws, size_t ws_size,
                              hipStream_t stream) {
  const float* vmodel  = (const float*)d_in[0];   // (8192, 2500) f32
  const float* wavelet = (const float*)d_in[1];   // (101,) f32
  float* out = (float*)d_out;                     // (8192, 2000) f32

  const size_t shmem = (size_t)SMEM_FLOATS * sizeof(float);   // ~137 KB (< 320 KB/WGP on CDNA5)
  (void)hipFuncSetAttribute((const void*)convmodel_wmma_kernel,
                            hipFuncAttributeMaxDynamicSharedMemorySize, (int)shmem);
  dim3 grid(B_ROWS / ROW_TILE);   // 512 blocks of 16 rows
  convmodel_wmma_kernel<<<grid, 256, shmem, stream>>>(vmodel, wavelet, out);
}